// GIN_65386582114733
// MI455X (gfx1250) — compile-verified
//
#include <hip/hip_runtime.h>
#include <hip/hip_bf16.h>

typedef __attribute__((ext_vector_type(2))) float v2f;
typedef __attribute__((ext_vector_type(8))) float v8f;

#define N_NODES   10000
#define NUM_EDGES 640000
#define NUM_GRAPHS 64
#define DIM_IN    128
#define DIM_H     256
#define DIM_OUT   64

// ---------------------------------------------------------------------------
// zero a float4-aligned region
// ---------------------------------------------------------------------------
__global__ void gin_zero_f4(float4* __restrict__ p, int n4) {
    int i = blockIdx.x * blockDim.x + threadIdx.x;
    int stride = gridDim.x * blockDim.x;
    float4 z = make_float4(0.f, 0.f, 0.f, 0.f);
    for (; i < n4; i += stride) p[i] = z;
}

// ---------------------------------------------------------------------------
// scatter-add: aggr[dst[e]][:] += x[src[e]][:]   (wave per edge, lanes = feat)
// ---------------------------------------------------------------------------
__global__ void gin_scatter_add(const float* __restrict__ x,
                                const int* __restrict__ src,
                                const int* __restrict__ dst,
                                float* __restrict__ aggr,
                                int E, int D4) {
    int tid   = blockIdx.x * blockDim.x + threadIdx.x;
    int lane  = threadIdx.x & 31;
    int wave  = tid >> 5;
    int nwave = (gridDim.x * blockDim.x) >> 5;
    const float4* x4 = (const float4*)x;
    for (int e = wave; e < E; e += nwave) {
        int s = __builtin_amdgcn_readfirstlane(src[e]);
        int d = __builtin_amdgcn_readfirstlane(dst[e]);
        for (int c = lane; c < D4; c += 32) {
            float4 v = x4[(long)s * D4 + c];
            float* a = aggr + ((long)d * D4 + c) * 4;
            atomicAdd(a + 0, v.x);
            atomicAdd(a + 1, v.y);
            atomicAdd(a + 2, v.z);
            atomicAdd(a + 3, v.w);
        }
    }
}

// ---------------------------------------------------------------------------
// t = (1 + eps) * h + aggr      (eps is a device scalar)
// ---------------------------------------------------------------------------
__global__ void gin_fuse_eps(const float* __restrict__ h,
                             const float* __restrict__ aggr,
                             const float* __restrict__ eps,
                             float* __restrict__ t, int n4) {
    float e = 1.0f + eps[0];
    const float4* h4 = (const float4*)h;
    const float4* a4 = (const float4*)aggr;
    float4* t4 = (float4*)t;
    int i = blockIdx.x * blockDim.x + threadIdx.x;
    int stride = gridDim.x * blockDim.x;
    for (; i < n4; i += stride) {
        float4 hv = h4[i];
        float4 av = a4[i];
        float4 r;
        r.x = fmaf(e, hv.x, av.x);
        r.y = fmaf(e, hv.y, av.y);
        r.z = fmaf(e, hv.z, av.z);
        r.w = fmaf(e, hv.w, av.w);
        t4[i] = r;
    }
}

// ---------------------------------------------------------------------------
// C[M,N] = act(A[M,K] @ W[K,N] + bias)  via V_WMMA_F32_16X16X4_F32
// grid = (M/16, N/(16*8)), block = 256 (8 waves), wave -> one 16x16 tile
// M, N multiples of 16; K multiple of 4. EXEC all ones (no divergence).
// ---------------------------------------------------------------------------
__global__ void gin_gemm_bias_relu(const float* __restrict__ A,
                                   const float* __restrict__ W,
                                   const float* __restrict__ bias,
                                   float* __restrict__ C,
                                   int M, int K, int Ncols, int do_relu) {
    int wave = threadIdx.x >> 5;
    int lane = threadIdx.x & 31;
    int half = lane >> 4;      // selects K pair of the 16x4 A fragment
    int r    = lane & 15;      // row (A) / col (B) within tile
    int m0 = blockIdx.x * 16;
    int n0 = (blockIdx.y * 8 + wave) * 16;

    const float* arow = A + (long)(m0 + r) * K;
    v8f acc = {};
#pragma unroll 4
    for (int k0 = 0; k0 < K; k0 += 4) {
        int k = k0 + 2 * half;
        float2 af = *(const float2*)(arow + k);
        v2f a; a.x = af.x; a.y = af.y;
        v2f b;
        b.x = W[(long)k * Ncols + n0 + r];
        b.y = W[(long)(k + 1) * Ncols + n0 + r];
        acc = __builtin_amdgcn_wmma_f32_16x16x4_f32(
            /*neg_a=*/false, a, /*neg_b=*/false, b,
            /*c_mod=*/(short)0, acc, /*reuse_a=*/false, /*reuse_b=*/false);
    }
    float bv = bias[n0 + r];
#pragma unroll
    for (int j = 0; j < 8; ++j) {
        int m = m0 + j + 8 * half;           // C layout: VGPR j -> row j (+8 for hi half)
        float v = acc[j] + bv;
        if (do_relu) v = fmaxf(v, 0.f);
        C[(long)m * Ncols + n0 + r] = v;
    }
}

// ---------------------------------------------------------------------------
// segment-sum pooling: pooled[batch[n]] += h[n]; counts[batch[n]] += 1
// ---------------------------------------------------------------------------
__global__ void gin_pool_accum(const float* __restrict__ h,
                               const int* __restrict__ batch,
                               float* __restrict__ pooled,
                               float* __restrict__ counts,
                               int n, int D4) {
    int tid   = blockIdx.x * blockDim.x + threadIdx.x;
    int lane  = threadIdx.x & 31;
    int wave  = tid >> 5;
    int nwave = (gridDim.x * blockDim.x) >> 5;
    const float4* h4 = (const float4*)h;
    for (int node = wave; node < n; node += nwave) {
        int g = __builtin_amdgcn_readfirstlane(batch[node]);
        for (int c = lane; c < D4; c += 32) {
            float4 v = h4[(long)node * D4 + c];
            float* p = pooled + ((long)g * D4 + c) * 4;
            atomicAdd(p + 0, v.x);
            atomicAdd(p + 1, v.y);
            atomicAdd(p + 2, v.z);
            atomicAdd(p + 3, v.w);
        }
        if (lane == 0) atomicAdd(&counts[g], 1.0f);
    }
}

// ---------------------------------------------------------------------------
// out[64,64] = (pooled/max(counts,1)) @ Wo[256,64] + bo
// one block, 512 threads = 16 waves = all 16 output tiles
// ---------------------------------------------------------------------------
__global__ void gin_head_gemm(const float* __restrict__ pooled,
                              const float* __restrict__ counts,
                              const float* __restrict__ Wo,
                              const float* __restrict__ bo,
                              float* __restrict__ out) {
    int wave = threadIdx.x >> 5;
    int lane = threadIdx.x & 31;
    int half = lane >> 4;
    int r    = lane & 15;
    int m0 = (wave >> 2) * 16;   // 4 M-tiles
    int n0 = (wave & 3) * 16;    // 4 N-tiles

    float inv = 1.0f / fmaxf(counts[m0 + r], 1.0f);   // per A-row mean scale
    const float* arow = pooled + (long)(m0 + r) * DIM_H;
    v8f acc = {};
#pragma unroll 4
    for (int k0 = 0; k0 < DIM_H; k0 += 4) {
        int k = k0 + 2 * half;
        float2 af = *(const float2*)(arow + k);
        v2f a; a.x = af.x * inv; a.y = af.y * inv;
        v2f b;
        b.x = Wo[(long)k * DIM_OUT + n0 + r];
        b.y = Wo[(long)(k + 1) * DIM_OUT + n0 + r];
        acc = __builtin_amdgcn_wmma_f32_16x16x4_f32(
            false, a, false, b, (short)0, acc, false, false);
    }
    float bv = bo[n0 + r];
#pragma unroll
    for (int j = 0; j < 8; ++j) {
        int m = m0 + j + 8 * half;
        out[(long)m * DIM_OUT + n0 + r] = acc[j] + bv;
    }
}

// ---------------------------------------------------------------------------
extern "C" void kernel_launch(void* const* d_in, const int* in_sizes, int n_in,
                              void* d_out, int out_size, void* d_ws, size_t ws_size,
                              hipStream_t stream) {
    const float* x     = (const float*)d_in[0];
    const int*   ei    = (const int*)d_in[1];
    const int*   src   = ei;               // edge_index[0]
    const int*   dst   = ei + NUM_EDGES;   // edge_index[1]
    const int*   batch = (const int*)d_in[2];
    const float* W1 = (const float*)d_in[3];
    const float* b1 = (const float*)d_in[4];
    const float* W2 = (const float*)d_in[5];
    const float* b2 = (const float*)d_in[6];
    const float* W3 = (const float*)d_in[7];
    const float* b3 = (const float*)d_in[8];
    const float* Wo = (const float*)d_in[9];
    const float* bo = (const float*)d_in[10];
    const float* eps = (const float*)d_in[11];
    float* out = (float*)d_out;

    const size_t SLAB = (size_t)N_NODES * DIM_H * sizeof(float); // 10.24 MB
    char* ws = (char*)d_ws;
    float* aggr   = (float*)(ws + 0 * SLAB);
    float* t      = (float*)(ws + 1 * SLAB);
    float* h1     = (float*)(ws + 2 * SLAB);
    float* h2     = (float*)(ws + 3 * SLAB);
    float* pooled = (float*)(ws + 4 * SLAB);
    float* counts = pooled + NUM_GRAPHS * DIM_H;   // contiguous after pooled

    const int BLK = 256;
    const int SCATTER_BLOCKS = 1280;  // 10240 waves over 640k edges

    // ---------- Layer 1 (D_IN=128 -> D_H=256) ----------
    {
        int n4 = N_NODES * DIM_IN / 4;
        gin_zero_f4<<<(n4 + BLK - 1) / BLK, BLK, 0, stream>>>((float4*)aggr, n4);
        gin_scatter_add<<<SCATTER_BLOCKS, BLK, 0, stream>>>(x, src, dst, aggr,
                                                            NUM_EDGES, DIM_IN / 4);
        gin_fuse_eps<<<(n4 + BLK - 1) / BLK, BLK, 0, stream>>>(x, aggr, eps + 0, t, n4);
        dim3 g(N_NODES / 16, DIM_H / 128);
        gin_gemm_bias_relu<<<g, BLK, 0, stream>>>(t, W1, b1, h1,
                                                  N_NODES, DIM_IN, DIM_H, 1);
    }
    // ---------- Layer 2 (256 -> 256) ----------
    {
        int n4 = N_NODES * DIM_H / 4;
        gin_zero_f4<<<(n4 + BLK - 1) / BLK, BLK, 0, stream>>>((float4*)aggr, n4);
        gin_scatter_add<<<SCATTER_BLOCKS, BLK, 0, stream>>>(h1, src, dst, aggr,
                                                            NUM_EDGES, DIM_H / 4);
        gin_fuse_eps<<<(n4 + BLK - 1) / BLK, BLK, 0, stream>>>(h1, aggr, eps + 1, t, n4);
        dim3 g(N_NODES / 16, DIM_H / 128);
        gin_gemm_bias_relu<<<g, BLK, 0, stream>>>(t, W2, b2, h2,
                                                  N_NODES, DIM_H, DIM_H, 1);
    }
    // ---------- Layer 3 (256 -> 256), output into h1 ----------
    {
        int n4 = N_NODES * DIM_H / 4;
        gin_zero_f4<<<(n4 + BLK - 1) / BLK, BLK, 0, stream>>>((float4*)aggr, n4);
        gin_scatter_add<<<SCATTER_BLOCKS, BLK, 0, stream>>>(h2, src, dst, aggr,
                                                            NUM_EDGES, DIM_H / 4);
        gin_fuse_eps<<<(n4 + BLK - 1) / BLK, BLK, 0, stream>>>(h2, aggr, eps + 2, t, n4);
        dim3 g(N_NODES / 16, DIM_H / 128);
        gin_gemm_bias_relu<<<g, BLK, 0, stream>>>(t, W3, b3, h1,
                                                  N_NODES, DIM_H, DIM_H, 1);
    }
    // ---------- Global mean pool + head GEMM ----------
    {
        int n4 = (NUM_GRAPHS * DIM_H + NUM_GRAPHS) / 4;  // pooled + counts
        gin_zero_f4<<<(n4 + BLK - 1) / BLK, BLK, 0, stream>>>((float4*)pooled, n4);
        gin_pool_accum<<<320, BLK, 0, stream>>>(h1, batch, pooled, counts,
                                                N_NODES, DIM_H / 4);
        gin_head_gemm<<<1, 512, 0, stream>>>(pooled, counts, Wo, bo, out);
    }
}